// RGCN_55422257988217
// MI455X (gfx1250) — compile-verified
//
#include <hip/hip_runtime.h>

#define NFRAME 100000
#define NFE    200000
#define FDIM   512
#define HIDD   128
#define NEDGE  1000000

typedef __attribute__((ext_vector_type(2))) float v2f;
typedef __attribute__((ext_vector_type(8))) float v8f;

static constexpr int KS   = 64;   // K-chunk for streaming GEMM
static constexpr int APAD = 4;    // pad: 16B-aligned rows + conflict-free column reads

// ---------------------------------------------------------------------------
// gfx1250 async global->LDS copy (ASYNCcnt-tracked), 16 bytes per lane.
// ---------------------------------------------------------------------------
__device__ __forceinline__ void async_ld16(unsigned int lds_off, const float* g)
{
    asm volatile("global_load_async_to_lds_b128 %0, %1, off"
                 :: "v"(lds_off), "v"(g) : "memory");
}
__device__ __forceinline__ void wait_async0() { asm volatile("s_wait_asynccnt 0x0" ::: "memory"); }
__device__ __forceinline__ void wait_async1() { asm volatile("s_wait_asynccnt 0x1" ::: "memory"); }
__device__ __forceinline__ void wait_async2() { asm volatile("s_wait_asynccnt 0x2" ::: "memory"); }

__device__ __forceinline__ unsigned int lds_off32(const void* p)
{
    return (unsigned int)(unsigned long long)p;   // low 32 bits = LDS offset
}

// ---------------------------------------------------------------------------
// Streaming WMMA GEMM (used for K=512 input linears, N=128):
// A tiles (16xKS) double-buffered in LDS via async copies; B streamed (L2-hot).
// ---------------------------------------------------------------------------
__global__ __launch_bounds__(256)
void wmma_gemm_stream(const float* __restrict__ A, const float* __restrict__ W,
                      const float* __restrict__ bias, float* __restrict__ out,
                      int M, int K, int N)
{
    __shared__ float As[2][16][KS + APAD];

    const int m0     = blockIdx.x * 16;
    const int wave   = threadIdx.x >> 5;
    const int lane   = threadIdx.x & 31;
    const int n_base = blockIdx.y * 128 + wave * 16;
    const int mrow   = lane & 15;
    const int khalf  = (lane >> 4) * 2;
    const int ncol   = n_base + (lane & 15);
    // staging map: thread t copies 16B: row = t>>4, 4-float col chunk = (t&15)*4
    const int sr = threadIdx.x >> 4;
    const int sc = (threadIdx.x & 15) * 4;

    v8f acc = {0.f, 0.f, 0.f, 0.f, 0.f, 0.f, 0.f, 0.f};

    // prologue: tile 0 in flight
    async_ld16(lds_off32(&As[0][sr][sc]), A + (size_t)(m0 + sr) * K + sc);

    const int nT = K / KS;
    for (int t = 0; t < nT; ++t) {
        const int k0 = t * KS;
        if (t + 1 < nT) {
            async_ld16(lds_off32(&As[(t + 1) & 1][sr][sc]),
                       A + (size_t)(m0 + sr) * K + (k0 + KS + sc));
            wait_async1();                 // tile t complete, t+1 may be in flight
        } else {
            wait_async0();
        }
        __syncthreads();

        const float (*as)[KS + APAD] = As[t & 1];
        #pragma unroll
        for (int kk = 0; kk < KS; kk += 4) {
            v2f a, b;
            a.x = as[mrow][kk + khalf + 0];
            a.y = as[mrow][kk + khalf + 1];
            const float* wp = W + (size_t)(k0 + kk + khalf) * N + ncol;
            b.x = wp[0];
            b.y = wp[N];
            acc = __builtin_amdgcn_wmma_f32_16x16x4_f32(
                false, a, false, b, (short)0, acc, false, false);
        }
        __syncthreads();
    }

    const int   mbase = m0 + ((lane >> 4) << 3);
    const float bval  = bias[ncol];
    #pragma unroll
    for (int r = 0; r < 8; ++r) {
        size_t idx = (size_t)(mbase + r) * N + ncol;
        out[idx] = acc[r] + bval;
    }
}

// ---------------------------------------------------------------------------
// K=128 WMMA GEMM with per-wave B resident in registers (32 v2f = 64 VGPRs),
// grid-stride over M tiles, A tiles async double-buffered in LDS.
// epilogue: v = rowscale*acc + bias ; [+= prev] ; [*0.5] ; [+res] ; [relu]
// ---------------------------------------------------------------------------
template<bool RSCALE, bool ACC, bool HALF, bool RELU, bool RES>
__global__ __launch_bounds__(256)
void wmma_gemm_k128(const float* __restrict__ A, const float* __restrict__ W,
                    const float* __restrict__ bias, const float* __restrict__ rowscale,
                    const float* __restrict__ resid, float* __restrict__ out,
                    int M, int N)
{
    constexpr int K = 128;
    __shared__ float As[2][16][K + APAD];

    const int wave   = threadIdx.x >> 5;
    const int lane   = threadIdx.x & 31;
    const int n_base = blockIdx.y * 128 + wave * 16;
    const int mrow   = lane & 15;
    const int khalf  = (lane >> 4) * 2;
    const int ncol   = n_base + (lane & 15);
    const int sr = threadIdx.x >> 4;
    const int sc = (threadIdx.x & 15) * 4;

    // preload all 32 B fragments for this wave's 16 columns (weights L2-hot)
    v2f Bf[32];
    #pragma unroll
    for (int ks = 0; ks < 32; ++ks) {
        const float* wp = W + (size_t)(ks * 4 + khalf) * N + ncol;
        Bf[ks].x = wp[0];
        Bf[ks].y = wp[N];
    }

    const int nTiles = M / 16;
    if ((int)blockIdx.x >= nTiles) return;

    // stage one 16x128 A tile: 2 async b128 chunks per thread
    auto issue = [&](int tile, int buf) {
        const int m0 = tile * 16;
        #pragma unroll
        for (int c = 0; c < 2; ++c) {
            async_ld16(lds_off32(&As[buf][sr][sc + c * 64]),
                       A + (size_t)(m0 + sr) * K + (sc + c * 64));
        }
    };

    int cur = 0;
    issue(blockIdx.x, 0);
    for (int tile = blockIdx.x; tile < nTiles; tile += gridDim.x) {
        const int nxt = tile + gridDim.x;
        if (nxt < nTiles) { issue(nxt, cur ^ 1); wait_async2(); }
        else              { wait_async0(); }
        __syncthreads();

        const float (*as)[K + APAD] = As[cur];
        v8f acc = {0.f, 0.f, 0.f, 0.f, 0.f, 0.f, 0.f, 0.f};
        #pragma unroll
        for (int ks = 0; ks < 32; ++ks) {
            v2f a;
            a.x = as[mrow][ks * 4 + khalf + 0];
            a.y = as[mrow][ks * 4 + khalf + 1];
            acc = __builtin_amdgcn_wmma_f32_16x16x4_f32(
                false, a, false, Bf[ks], (short)0, acc, false, false);
        }

        const int   m0    = tile * 16;
        const int   mbase = m0 + ((lane >> 4) << 3);
        const float bval  = bias[ncol];
        #pragma unroll
        for (int r = 0; r < 8; ++r) {
            const int row = mbase + r;
            size_t idx = (size_t)row * N + ncol;
            float v = acc[r];
            if (RSCALE) v *= rowscale[row];     // diag(nd) * (A@W)
            v += bval;
            if (ACC)  v += out[idx];
            if (HALF) v *= 0.5f;
            if (RES)  v += resid[idx];
            if (RELU) v = fmaxf(v, 0.f);
            out[idx] = v;
        }
        __syncthreads();
        cur ^= 1;
    }
}

// ---------------------------------------------------------------------------
// Edge scatter: one wave per edge; lane moves float4 (32 lanes x 4 = 128 feats)
// agg[dst] += h[src] * ns[src]   (f32 atomics, L2-resident targets)
// ---------------------------------------------------------------------------
__global__ __launch_bounds__(256)
void scatter_kernel(const float* __restrict__ h, const float* __restrict__ ns,
                    const int* __restrict__ src, const int* __restrict__ dst,
                    float* __restrict__ agg, int nE)
{
    const int lane = threadIdx.x & 31;
    const int wid  = (blockIdx.x * blockDim.x + threadIdx.x) >> 5;
    const int nW   = (gridDim.x * blockDim.x) >> 5;
    for (int e = wid; e < nE; e += nW) {
        const int s = src[e], d = dst[e];
        const float sc = ns[s];
        const float4 hv = *reinterpret_cast<const float4*>(h + (size_t)s * HIDD + lane * 4);
        float* ap = agg + (size_t)d * HIDD + lane * 4;
        atomicAdd(ap + 0, hv.x * sc);
        atomicAdd(ap + 1, hv.y * sc);
        atomicAdd(ap + 2, hv.z * sc);
        atomicAdd(ap + 3, hv.w * sc);
    }
}

__global__ void degree_kernel(const int* __restrict__ src, const int* __restrict__ dst,
                              float* __restrict__ deg_out, float* __restrict__ deg_in, int nE)
{
    int i = blockIdx.x * blockDim.x + threadIdx.x;
    if (i < nE) {
        atomicAdd(&deg_out[src[i]], 1.f);
        atomicAdd(&deg_in[dst[i]], 1.f);
    }
}

__global__ void normalize_kernel(float* __restrict__ deg, int n)
{
    int i = blockIdx.x * blockDim.x + threadIdx.x;
    if (i < n) {
        float d = deg[i];
        deg[i] = d > 0.f ? rsqrtf(fmaxf(d, 1.f)) : 0.f;
    }
}

// ---------------------------------------------------------------------------
extern "C" void kernel_launch(void* const* d_in, const int* in_sizes, int n_in,
                              void* d_out, int out_size, void* d_ws, size_t ws_size,
                              hipStream_t stream)
{
    (void)in_sizes; (void)n_in; (void)out_size; (void)ws_size;

    const float* frame   = (const float*)d_in[0];
    const float* fe      = (const float*)d_in[1];
    const int* s_frfr = (const int*)d_in[2];
    const int* t_frfr = (const int*)d_in[3];
    const int* s_frfe = (const int*)d_in[4];
    const int* t_frfe = (const int*)d_in[5];
    const int* s_fefr = (const int*)d_in[6];
    const int* t_fefr = (const int*)d_in[7];
    const int* s_fefe = (const int*)d_in[8];
    const int* t_fefe = (const int*)d_in[9];
    const float* W_fr_lin = (const float*)d_in[10];
    const float* b_fr_lin = (const float*)d_in[11];
    const float* W_fe_lin = (const float*)d_in[12];
    const float* b_fe_lin = (const float*)d_in[13];
    const float* W_frfr = (const float*)d_in[14];
    const float* b_frfr = (const float*)d_in[15];
    const float* W_frfe = (const float*)d_in[16];
    const float* b_frfe = (const float*)d_in[17];
    const float* W_fefr = (const float*)d_in[18];
    const float* b_fefr = (const float*)d_in[19];
    const float* W_fefe = (const float*)d_in[20];
    const float* b_fefe = (const float*)d_in[21];
    const float* W_fr_fc = (const float*)d_in[22];
    const float* b_fr_fc = (const float*)d_in[23];
    const float* W_fe_fc = (const float*)d_in[24];
    const float* b_fe_fc = (const float*)d_in[25];

    // output layout: fr_h [NF,512] | fe_h [NE,512] | h_fr [NF,128] | h_fe [NE,128]
    float* out    = (float*)d_out;
    float* o_frh  = out;
    float* o_feh  = o_frh + (size_t)NFRAME * FDIM;
    float* o_hfr  = o_feh + (size_t)NFE * FDIM;
    float* o_hfe  = o_hfr + (size_t)NFRAME * HIDD;

    // workspace carve-up
    float* ws = (float*)d_ws;
    size_t off = 0;
    auto wsa = [&](size_t n) { float* p = ws + off; off += n; return p; };
    float* x_fr  = wsa((size_t)NFRAME * HIDD);
    float* x_fe  = wsa((size_t)NFE * HIDD);
    float* h1_fr = wsa((size_t)NFRAME * HIDD);
    float* h1_fe = wsa((size_t)NFE * HIDD);
    float* agg_base = ws + off;
    float* agg_fr_a = wsa((size_t)NFRAME * HIDD);   // frfr -> fr
    float* agg_fr_b = wsa((size_t)NFRAME * HIDD);   // fefr -> fr
    float* agg_fe_a = wsa((size_t)NFE * HIDD);      // frfe -> fe
    float* agg_fe_b = wsa((size_t)NFE * HIDD);      // fefe -> fe
    const size_t aggBytes = (size_t)(2 * NFRAME + 2 * NFE) * HIDD * sizeof(float);
    float* norm_base = ws + off;
    float* ns_frfr = wsa(NFRAME); float* nd_frfr = wsa(NFRAME);
    float* ns_frfe = wsa(NFRAME); float* nd_frfe = wsa(NFE);
    float* ns_fefr = wsa(NFE);    float* nd_fefr = wsa(NFRAME);
    float* ns_fefe = wsa(NFE);    float* nd_fefe = wsa(NFE);
    const size_t normCount = (size_t)4 * NFRAME + (size_t)4 * NFE;

    // ---- degrees & norms (edges identical in both layers: compute once)
    hipMemsetAsync(norm_base, 0, normCount * sizeof(float), stream);
    const int degBlocks = (NEDGE + 255) / 256;
    degree_kernel<<<degBlocks, 256, 0, stream>>>(s_frfr, t_frfr, ns_frfr, nd_frfr, NEDGE);
    degree_kernel<<<degBlocks, 256, 0, stream>>>(s_frfe, t_frfe, ns_frfe, nd_frfe, NEDGE);
    degree_kernel<<<degBlocks, 256, 0, stream>>>(s_fefr, t_fefr, ns_fefr, nd_fefr, NEDGE);
    degree_kernel<<<degBlocks, 256, 0, stream>>>(s_fefe, t_fefe, ns_fefe, nd_fefe, NEDGE);
    normalize_kernel<<<(int)((normCount + 255) / 256), 256, 0, stream>>>(norm_base, (int)normCount);

    // ---- input linear projections (K=512 -> 128), streaming async-LDS GEMM
    wmma_gemm_stream<<<dim3(NFRAME / 16, 1), 256, 0, stream>>>(
        frame, W_fr_lin, b_fr_lin, x_fr, NFRAME, FDIM, HIDD);
    wmma_gemm_stream<<<dim3(NFE / 16, 1), 256, 0, stream>>>(
        fe, W_fe_lin, b_fe_lin, x_fe, NFE, FDIM, HIDD);

    const int SB = 4096;   // scatter grid (32k waves, grid-stride over 1M edges)
    const int GB = 2048;   // k128 GEMM grid-stride blocks (amortizes B preload)

    // ---- layer 1: h1 = relu(0.5 * (conv_a + conv_b))
    hipMemsetAsync(agg_base, 0, aggBytes, stream);
    scatter_kernel<<<SB, 256, 0, stream>>>(x_fr, ns_frfr, s_frfr, t_frfr, agg_fr_a, NEDGE);
    scatter_kernel<<<SB, 256, 0, stream>>>(x_fe, ns_fefr, s_fefr, t_fefr, agg_fr_b, NEDGE);
    scatter_kernel<<<SB, 256, 0, stream>>>(x_fr, ns_frfe, s_frfe, t_frfe, agg_fe_a, NEDGE);
    scatter_kernel<<<SB, 256, 0, stream>>>(x_fe, ns_fefe, s_fefe, t_fefe, agg_fe_b, NEDGE);
    wmma_gemm_k128<true,false,false,false,false><<<dim3(GB, 1), 256, 0, stream>>>(
        agg_fr_a, W_frfr, b_frfr, nd_frfr, nullptr, h1_fr, NFRAME, HIDD);
    wmma_gemm_k128<true,true,true,true,false><<<dim3(GB, 1), 256, 0, stream>>>(
        agg_fr_b, W_fefr, b_fefr, nd_fefr, nullptr, h1_fr, NFRAME, HIDD);
    wmma_gemm_k128<true,false,false,false,false><<<dim3(GB, 1), 256, 0, stream>>>(
        agg_fe_a, W_frfe, b_frfe, nd_frfe, nullptr, h1_fe, NFE, HIDD);
    wmma_gemm_k128<true,true,true,true,false><<<dim3(GB, 1), 256, 0, stream>>>(
        agg_fe_b, W_fefe, b_fefe, nd_fefe, nullptr, h1_fe, NFE, HIDD);

    // ---- layer 2: h = 0.5 * (conv_a + conv_b) + x   (into d_out h regions)
    hipMemsetAsync(agg_base, 0, aggBytes, stream);
    scatter_kernel<<<SB, 256, 0, stream>>>(h1_fr, ns_frfr, s_frfr, t_frfr, agg_fr_a, NEDGE);
    scatter_kernel<<<SB, 256, 0, stream>>>(h1_fe, ns_fefr, s_fefr, t_fefr, agg_fr_b, NEDGE);
    scatter_kernel<<<SB, 256, 0, stream>>>(h1_fr, ns_frfe, s_frfe, t_frfe, agg_fe_a, NEDGE);
    scatter_kernel<<<SB, 256, 0, stream>>>(h1_fe, ns_fefe, s_fefe, t_fefe, agg_fe_b, NEDGE);
    wmma_gemm_k128<true,false,false,false,false><<<dim3(GB, 1), 256, 0, stream>>>(
        agg_fr_a, W_frfr, b_frfr, nd_frfr, nullptr, o_hfr, NFRAME, HIDD);
    wmma_gemm_k128<true,true,true,false,true><<<dim3(GB, 1), 256, 0, stream>>>(
        agg_fr_b, W_fefr, b_fefr, nd_fefr, x_fr, o_hfr, NFRAME, HIDD);
    wmma_gemm_k128<true,false,false,false,false><<<dim3(GB, 1), 256, 0, stream>>>(
        agg_fe_a, W_frfe, b_frfe, nd_frfe, nullptr, o_hfe, NFE, HIDD);
    wmma_gemm_k128<true,true,true,false,true><<<dim3(GB, 1), 256, 0, stream>>>(
        agg_fe_b, W_fefe, b_fefe, nd_fefe, x_fe, o_hfe, NFE, HIDD);

    // ---- final FC (K=128 -> 512), B-resident GEMM, grid.y covers N=512
    wmma_gemm_k128<false,false,false,false,false><<<dim3(GB, 4), 256, 0, stream>>>(
        o_hfr, W_fr_fc, b_fr_fc, nullptr, nullptr, o_frh, NFRAME, FDIM);
    wmma_gemm_k128<false,false,false,false,false><<<dim3(GB, 4), 256, 0, stream>>>(
        o_hfe, W_fe_fc, b_fe_fc, nullptr, nullptr, o_feh, NFE, FDIM);
}